// LinearImputer_29815662968983
// MI455X (gfx1250) — compile-verified
//
#include <hip/hip_runtime.h>

// Problem constants from the reference: x_masked is [B, T, D] float32.
#define BB 32
#define TT 4096
#define DD 256
#define CHUNK 128
#define NC (TT / CHUNK)
#define SUB 32                  // t-rows per TDM sub-tile
#define NSUB (CHUNK / SUB)

typedef unsigned int v4u __attribute__((ext_vector_type(4)));
typedef int          v8i __attribute__((ext_vector_type(8)));
typedef int          v4i __attribute__((ext_vector_type(4)));

// ---------------------------------------------------------------------------
// TDM: DMA a [SUB rows x DD cols] fp32 tile (row stride DD elems) into LDS.
// Descriptor packed per CDNA5 ISA 8.3/8.4 (D# group0 128b, group1 256b).
// ---------------------------------------------------------------------------
__device__ __forceinline__ void tdm_load_tile(const float* gsrc, unsigned ldsByteAddr) {
    unsigned long long ga = (unsigned long long)(uintptr_t)gsrc;

    v4u g0;
    g0.x = 1u;                                       // count=1, is_restore=0, gather off
    g0.y = ldsByteAddr;                              // lds_addr [63:32]
    g0.z = (unsigned)(ga & 0xFFFFFFFFu);             // global_addr [95:64]
    g0.w = (unsigned)((ga >> 32) & 0x01FFFFFFu)      // global_addr [120:96]
         | (2u << 30);                               // type=2 ("image")

    v8i g1;
    g1[0] = (int)(2u << 16);                         // data_size=2 (4B); mask/pad/iter=0
    g1[1] = (int)(((unsigned)DD & 0xFFFFu) << 16);   // abar=0 | tensor_dim0[15:0]
    g1[2] = (int)((((unsigned)DD >> 16) & 0xFFFFu)   // tensor_dim0[31:16]
         | (((unsigned)SUB & 0xFFFFu) << 16));       // tensor_dim1[15:0]
    g1[3] = (int)((((unsigned)SUB >> 16) & 0xFFFFu)  // tensor_dim1[31:16]
         | (((unsigned)DD & 0xFFFFu) << 16));        // tile_dim0
    g1[4] = (int)((unsigned)SUB & 0xFFFFu);          // tile_dim1 | tile_dim2=0
    g1[5] = DD;                                      // tensor_dim0_stride[31:0]
    g1[6] = 0;                                       // stride[47:32] | dim1_stride lo
    g1[7] = 0;

    v4i z4 = {0, 0, 0, 0};
#if defined(__clang_major__) && (__clang_major__ >= 23)
    v8i z8 = {0, 0, 0, 0, 0, 0, 0, 0};
    __builtin_amdgcn_tensor_load_to_lds(g0, g1, z4, z4, z8, 0);
#else
    __builtin_amdgcn_tensor_load_to_lds(g0, g1, z4, z4, 0);
#endif
}

// ---------------------------------------------------------------------------
// Fill out[lo..hi) of a zero-run with linspace(xp, xn, L) semantics.
// prev = last valid index before run (global t), nxt = first valid after
// (global t, == TT if none). Boundary-touching runs -> zeros.
// ---------------------------------------------------------------------------
__device__ __forceinline__ void flush_run(float* __restrict__ out, size_t rowBase,
                                          int lo, int hi, int prev, float xp,
                                          int nxt, float xn) {
    if (lo >= hi) return;
    if (prev < 0 || nxt >= TT) {
        for (int t = lo; t < hi; ++t) out[rowBase + (size_t)t * DD] = 0.0f;
    } else {
        int denomI = nxt - prev - 2;            // run_len - 1
        if (denomI < 1) denomI = 1;             // L == 1 -> frac 0 -> xp
        float fd = (float)denomI;
        float dx = xn - xp;
        for (int t = lo; t < hi; ++t) {
            float frac = (float)(t - prev - 1) / fd;
            out[rowBase + (size_t)t * DD] = xp + dx * frac;
        }
    }
}

// ---------------------------------------------------------------------------
// Phase 1: per (b, chunk, d) find first/last valid element in the chunk.
// ---------------------------------------------------------------------------
__global__ void seg_scan_kernel(const float* __restrict__ x, float4* __restrict__ seg) {
    int c = blockIdx.x % NC;
    int b = blockIdx.x / NC;
    int d = threadIdx.x;
    size_t base = ((size_t)b * TT + (size_t)c * CHUNK) * DD + (size_t)d;

    int firstIdx = -1; float firstVal = 0.0f;
    for (int i = 0; i < CHUNK; ++i) {
        float v = x[base + (size_t)i * DD];
        if (v != 0.0f) { firstIdx = c * CHUNK + i; firstVal = v; break; }
    }
    int lastIdx = -1; float lastVal = 0.0f;
    for (int i = CHUNK - 1; i >= 0; --i) {
        float v = x[base + (size_t)i * DD];
        if (v != 0.0f) { lastIdx = c * CHUNK + i; lastVal = v; break; }
    }
    float4 s;
    s.x = __int_as_float(firstIdx); s.y = firstVal;
    s.z = __int_as_float(lastIdx);  s.w = lastVal;
    seg[((size_t)b * NC + c) * DD + d] = s;
}

// ---------------------------------------------------------------------------
// Phase 2: per (b, d), carry last-valid forward / first-valid backward over
// chunks, producing per-chunk global boundary info.
// ---------------------------------------------------------------------------
__global__ void bounds_kernel(const float4* __restrict__ seg,
                              float2* __restrict__ prevA,
                              float2* __restrict__ nextA) {
    int b = blockIdx.x;
    int d = threadIdx.x;

    int prevIdx = -1; float prevVal = 0.0f;
    for (int c = 0; c < NC; ++c) {
        size_t i = ((size_t)b * NC + c) * DD + d;
        prevA[i] = make_float2(__int_as_float(prevIdx), prevVal);
        float4 s = seg[i];
        int lastIdx = __float_as_int(s.z);
        if (lastIdx >= 0) { prevIdx = lastIdx; prevVal = s.w; }
    }
    int nextIdx = TT; float nextVal = 0.0f;
    for (int c = NC - 1; c >= 0; --c) {
        size_t i = ((size_t)b * NC + c) * DD + d;
        nextA[i] = make_float2(__int_as_float(nextIdx), nextVal);
        float4 s = seg[i];
        int firstIdx = __float_as_int(s.x);
        if (firstIdx >= 0) { nextIdx = firstIdx; nextVal = s.y; }
    }
}

// ---------------------------------------------------------------------------
// Phase 3 (TDM path): one block per (b, chunk). Input tile is DMA'd into LDS
// with TENSOR_LOAD_TO_LDS, double-buffered in SUB-row sub-tiles. Lane = d;
// LDS reads are conflict-free. Outputs stream directly to global.
// ---------------------------------------------------------------------------
__global__ void interp_tdm_kernel(const float* __restrict__ x, float* __restrict__ out,
                                  const float2* __restrict__ prevA,
                                  const float2* __restrict__ nextA) {
    __shared__ float tile[2][SUB * DD];            // 2 x 32 KiB

    int c = blockIdx.x % NC;
    int b = blockIdx.x / NC;
    int d = threadIdx.x;
    int t0 = c * CHUNK;
    int t1 = t0 + CHUNK;
    size_t rowBase = (size_t)b * TT * DD + (size_t)d;
    const float* gbase = x + ((size_t)b * TT + (size_t)t0) * DD;

    float2 p = prevA[((size_t)b * NC + c) * DD + d];
    int prev = __float_as_int(p.x);
    float xp = p.y;

    bool dmaWave = (threadIdx.x < 32);             // wave 0 drives the TDM
    unsigned lds0 = (unsigned)(uintptr_t)(&tile[0][0]);
    unsigned lds1 = (unsigned)(uintptr_t)(&tile[1][0]);

    if (dmaWave) tdm_load_tile(gbase, lds0);

    for (int s = 0; s < NSUB; ++s) {
        int cur = s & 1;
        if (dmaWave) {
            if (s + 1 < NSUB) {
                tdm_load_tile(gbase + (size_t)(s + 1) * SUB * DD, (s + 1) & 1 ? lds1 : lds0);
                __builtin_amdgcn_s_wait_tensorcnt(1);  // sub-tile s resident (in-order)
            } else {
                __builtin_amdgcn_s_wait_tensorcnt(0);
            }
        }
        __syncthreads();                            // tile[cur] visible to all waves

        int ts0 = t0 + s * SUB;
        const float* lt = &tile[cur][0];
#pragma unroll 4
        for (int k = 0; k < SUB; ++k) {
            int t = ts0 + k;
            float v = lt[k * DD + d];
            if (v != 0.0f) {
                int lo = prev + 1; if (lo < t0) lo = t0;
                flush_run(out, rowBase, lo, t, prev, xp, t, v);
                out[rowBase + (size_t)t * DD] = v;
                prev = t; xp = v;
            }
        }
        __syncthreads();                            // all waves done with tile[cur]
    }

    // Trailing run extending to (or past) the chunk end.
    if (prev < t1 - 1) {
        float2 n = nextA[((size_t)b * NC + c) * DD + d];
        int nxt = __float_as_int(n.x);
        float xn = n.y;
        int lo = prev + 1; if (lo < t0) lo = t0;
        flush_run(out, rowBase, lo, t1, prev, xp, nxt, xn);
    }
}

// ---------------------------------------------------------------------------
// Workspace-free fallback: direct-load whole-row scan (one block per b).
// ---------------------------------------------------------------------------
__global__ void interp_row_kernel(const float* __restrict__ x, float* __restrict__ out) {
    int b = blockIdx.x;
    int d = threadIdx.x;
    size_t rowBase = (size_t)b * TT * DD + (size_t)d;

    int prev = -1; float xp = 0.0f;
    for (int tb = 0; tb < TT; tb += 8) {
        {
            int pf = tb + 16;
            if (pf > TT - 8) pf = TT - 8;
            __builtin_prefetch(&x[rowBase + (size_t)pf * DD], 0, 1);
        }
        float buf[8];
#pragma unroll
        for (int k = 0; k < 8; ++k) buf[k] = x[rowBase + (size_t)(tb + k) * DD];
#pragma unroll
        for (int k = 0; k < 8; ++k) {
            int t = tb + k;
            float v = buf[k];
            if (v != 0.0f) {
                flush_run(out, rowBase, prev + 1, t, prev, xp, t, v);
                out[rowBase + (size_t)t * DD] = v;
                prev = t; xp = v;
            }
        }
    }
    if (prev < TT - 1) {
        flush_run(out, rowBase, prev + 1, TT, prev, xp, TT, 0.0f);
    }
}

extern "C" void kernel_launch(void* const* d_in, const int* in_sizes, int n_in,
                              void* d_out, int out_size, void* d_ws, size_t ws_size,
                              hipStream_t stream) {
    (void)in_sizes; (void)n_in; (void)out_size;
    const float* x = (const float*)d_in[0];
    float* out = (float*)d_out;

    const size_t nSeg = (size_t)BB * NC * DD;
    const size_t need = nSeg * (sizeof(float4) + 2 * sizeof(float2));  // 8 MiB

    if (ws_size >= need && d_ws != nullptr) {
        float4* seg   = (float4*)d_ws;
        float2* prevA = (float2*)((char*)d_ws + nSeg * sizeof(float4));
        float2* nextA = prevA + nSeg;

        seg_scan_kernel<<<BB * NC, DD, 0, stream>>>(x, seg);
        bounds_kernel<<<BB, DD, 0, stream>>>(seg, prevA, nextA);
        interp_tdm_kernel<<<BB * NC, DD, 0, stream>>>(x, out, prevA, nextA);
    } else {
        interp_row_kernel<<<BB, DD, 0, stream>>>(x, out);
    }
}